// WW_Attention_15753940042188
// MI455X (gfx1250) — compile-verified
//
#include <hip/hip_runtime.h>

typedef float v2f __attribute__((ext_vector_type(2)));
typedef float v8f __attribute__((ext_vector_type(8)));

#define Hd    768
#define Ld    4096
#define Bd    32
#define NROWS (Bd * Ld)
#define NEGV  (-1e30f)

// ---------------------------------------------------------------------------
// Pass 1: s[R] = words[R,:]·w1 ; t[R] = words[R,:]·w2  via V_WMMA_F32_16X16X4_F32
// Each wave handles 16 consecutive rows. A = 16x4 tile of words (f32),
// B = 4x16 tile with col0 = w1 chunk, col1 = w2 chunk, cols 2..15 = 0.
// D columns 0/1 accumulate s/t for the 16 rows. Memory-bound: 402 MB streamed.
// ---------------------------------------------------------------------------
__global__ __launch_bounds__(256) void ww_pass1_gemv_wmma(
    const float* __restrict__ words,   // [NROWS, Hd]
    const float* __restrict__ W,       // [2*Hd]  (w1 | w2)
    float* __restrict__ S,             // [NROWS]
    float* __restrict__ T)             // [NROWS]
{
    const int lane    = threadIdx.x & 31;
    const int wave    = threadIdx.x >> 5;
    const int rowBase = blockIdx.x * 128 + wave * 16;
    const int m       = lane & 15;   // A: row index; B: column index
    const int half    = lane >> 4;   // K split: lanes 16-31 hold K+2

    // A fragment source: row m of this wave's tile, K offset half*2
    const float* aptr = words + (size_t)(rowBase + m) * Hd + half * 2;
    // B fragment source: column 0 reads w1, column 1 reads w2, others zeroed
    const float* wptr = W + ((m == 1) ? Hd : 0) + half * 2;
    const float  sel  = (m < 2) ? 1.0f : 0.0f;

    v8f c0 = {};
    v8f c1 = {};

#pragma unroll 2
    for (int k = 0; k < Hd; k += 8) {
        // A: lane holds A[m][k + half*2 + {0,1}] in {x,y}
        v2f a0 = *(const v2f*)(aptr + k);
        v2f a1 = *(const v2f*)(aptr + k + 4);
        // B: lane holds B[k + half*2 + {0,1}][m] in {x,y}
        v2f b0, b1;
        b0.x = sel * wptr[k + 0];
        b0.y = sel * wptr[k + 1];
        b1.x = sel * wptr[k + 4];
        b1.y = sel * wptr[k + 5];
        // Two independent accumulator chains to hide WMMA latency
        c0 = __builtin_amdgcn_wmma_f32_16x16x4_f32(
                false, a0, false, b0, (short)0, c0, false, false);
        c1 = __builtin_amdgcn_wmma_f32_16x16x4_f32(
                false, a1, false, b1, (short)0, c1, false, false);
    }

    v8f c;
#pragma unroll
    for (int j = 0; j < 8; ++j) c[j] = c0[j] + c1[j];

    // D layout: VGPR j, lanes 0-15 = row M=j, lanes 16-31 = row M=j+8; N = lane&15.
    // s[row] lives in lanes 0/16 (N=0), t[row] in lanes 1/17 (N=1).
    if (m < 2) {
        float* dst = (m == 0) ? S : T;
#pragma unroll
        for (int j = 0; j < 8; ++j) dst[rowBase + half * 8 + j] = c[j];
    }
}

// ---------------------------------------------------------------------------
// Pass 2: out[b] = max(0, bias + max_l ( s[l] + max_{k=1..5, l+k<L} t[l+k] ))
// relu+global-max collapses to a 0-clamp of the global max; NEG padding drops out.
// Traffic ~1 MB total: negligible.
// ---------------------------------------------------------------------------
__global__ __launch_bounds__(256) void ww_pass2_window_max(
    const float* __restrict__ S,
    const float* __restrict__ T,
    const float* __restrict__ bias,
    float* __restrict__ out)
{
    __shared__ float red[256];
    const int b = blockIdx.x;
    const float* s = S + b * Ld;
    const float* t = T + b * Ld;

    float mx = NEGV;
    for (int l = threadIdx.x; l < Ld; l += 256) {
        float mt = NEGV;
#pragma unroll
        for (int k = 1; k <= 5; ++k) {
            float tv = (l + k < Ld) ? t[l + k] : NEGV;
            mt = fmaxf(mt, tv);
        }
        mx = fmaxf(mx, s[l] + mt);
    }
    red[threadIdx.x] = mx;
    __syncthreads();
    for (int off = 128; off > 0; off >>= 1) {
        if (threadIdx.x < off)
            red[threadIdx.x] = fmaxf(red[threadIdx.x], red[threadIdx.x + off]);
        __syncthreads();
    }
    if (threadIdx.x == 0) out[b] = fmaxf(0.0f, red[0] + bias[0]);
}

// ---------------------------------------------------------------------------
extern "C" void kernel_launch(void* const* d_in, const int* in_sizes, int n_in,
                              void* d_out, int out_size, void* d_ws, size_t ws_size,
                              hipStream_t stream)
{
    const float* words = (const float*)d_in[0];
    // d_in[1] = diff : unused by the reference
    const float* W     = (const float*)d_in[2];
    const float* bias  = (const float*)d_in[3];
    float*       out   = (float*)d_out;

    float* S = (float*)d_ws;          // NROWS floats
    float* T = S + NROWS;             // NROWS floats  (1 MB total)

    ww_pass1_gemv_wmma<<<NROWS / 128, 256, 0, stream>>>(words, W, S, T);
    ww_pass2_window_max<<<Bd, 256, 0, stream>>>(S, T, bias, out);
}